// EdgeLoss_39290360824757
// MI455X (gfx1250) — compile-verified
//
#include <hip/hip_runtime.h>
#include <stdint.h>

#define HW       512
#define TILE_W   64
#define TILE_H   16
#define HALO_W   66                  // TILE_W + 2
#define HALO_H   18                  // TILE_H + 2
#define NCELL    (HALO_W * HALO_H)   // 1188 halo cells per tensor

#if defined(__has_builtin)
#if __has_builtin(__builtin_amdgcn_tensor_load_to_lds)
#define HAVE_TDM 1
#endif
#endif
#ifndef HAVE_TDM
#define HAVE_TDM 0
#endif

typedef unsigned int u32x4 __attribute__((ext_vector_type(4)));
typedef int          i32x4 __attribute__((ext_vector_type(4)));
typedef int          i32x8 __attribute__((ext_vector_type(8)));

#if HAVE_TDM
// Issue one TDM descriptor: copy a HALO_W x HALO_H fp32 tile (row stride HW)
// starting at gsrc into LDS at lds_byte_addr. rem_w/rem_h = in-bounds extent
// from the tile start (interior tiles only, so no OOB zero-fill is needed).
__device__ __forceinline__ void tdm_load_tile(const float* gsrc,
                                              uint32_t lds_byte_addr,
                                              int rem_w, int rem_h)
{
    const uint64_t ga = (uint64_t)(uintptr_t)gsrc;
    u32x4 g0;
    g0.x = 1u;                                      // count=1, user load descriptor
    g0.y = lds_byte_addr;                           // LDS destination (bytes)
    g0.z = (uint32_t)(ga & 0xFFFFFFFFu);            // global_addr[31:0]
    g0.w = (uint32_t)((ga >> 32) & 0x01FFFFFFu)     // global_addr[56:32]
         | 0x80000000u;                             // type=2 ("image")
    i32x8 g1;
    g1[0] = 0x00020000;                             // wg_mask=0, data_size=4B
    g1[1] = (int)(((uint32_t)rem_w & 0xFFFFu) << 16);   // tensor_dim0[15:0]
    g1[2] = (int)(((uint32_t)rem_h & 0xFFFFu) << 16);   // dim0 hi=0 | tensor_dim1[15:0]
    g1[3] = (int)((uint32_t)HALO_W << 16);          // dim1 hi=0 | tile_dim0=66
    g1[4] = HALO_H;                                 // tile_dim1=18, tile_dim2=0
    g1[5] = HW;                                     // tensor_dim0_stride[31:0]=512
    g1[6] = 0;                                      // stride hi, dim1_stride lo
    g1[7] = 0;                                      // dim1_stride hi
    const i32x4 gz = {0, 0, 0, 0};
#if __clang_major__ >= 23
    const i32x8 gz8 = {0, 0, 0, 0, 0, 0, 0, 0};
    __builtin_amdgcn_tensor_load_to_lds(g0, g1, gz, gz, gz8, 0);
#else
    __builtin_amdgcn_tensor_load_to_lds(g0, g1, gz, gz, 0);
#endif
}
#endif // HAVE_TDM

// One workgroup = one 64x16 output tile of one 512x512 image.
// grid.x = n_images * 256  (8 tiles in x, 32 tiles in y)
__global__ __launch_bounds__(256) void sobel_edge_loss_kernel(
    const float* __restrict__ pred, const float* __restrict__ target,
    float* __restrict__ partials)
{
    __shared__ float sP[NCELL];
    __shared__ float sT[NCELL];
    __shared__ float sred[256];

    const int tid = threadIdx.x;
    const int wg  = blockIdx.x;
    const int img = wg >> 8;
    const int tin = wg & 255;
    const int ty  = tin >> 3;          // 0..31
    const int tx  = tin & 7;           // 0..7
    const int x0  = tx * TILE_W;
    const int y0  = ty * TILE_H;
    const size_t base = (size_t)img * (HW * HW);

    // Tile whose full halo is in-bounds -> TDM (or async) fast path.
    const bool interior = (tx >= 1) && (tx <= 6) && (ty >= 1) && (ty <= 30);

    if (interior) {
#if HAVE_TDM
        // Scalar (SGPR) branch so exactly ONE wave issues the DMAs:
        // TDM ignores EXEC, so an EXEC-masked branch is not sufficient.
        const int wave = __builtin_amdgcn_readfirstlane(tid) >> 5;
        if (wave == 0) {
            const size_t off = base + (size_t)(y0 - 1) * HW + (size_t)(x0 - 1);
            tdm_load_tile(pred + off,   (uint32_t)(uintptr_t)&sP[0], 513 - x0, 513 - y0);
            tdm_load_tile(target + off, (uint32_t)(uintptr_t)&sT[0], 513 - x0, 513 - y0);
            __builtin_amdgcn_s_wait_tensorcnt(0);
        }
#else
        // Fallback: per-lane CDNA5 async global->LDS copies (ASYNCcnt).
        for (int it = 0; it < 5; ++it) {
            const int i = tid + it * 256;
            if (i < NCELL) {
                const int hr = i / HALO_W;
                const int hc = i - hr * HALO_W;
                const size_t g = base + (size_t)(y0 - 1 + hr) * HW + (size_t)(x0 - 1 + hc);
                const float* gp = pred + g;
                const float* gt = target + g;
                uint32_t lp = (uint32_t)(uintptr_t)&sP[i];
                uint32_t lt = (uint32_t)(uintptr_t)&sT[i];
                asm volatile("global_load_async_to_lds_b32 %0, %1, off"
                             :: "v"(lp), "v"(gp) : "memory");
                asm volatile("global_load_async_to_lds_b32 %0, %1, off"
                             :: "v"(lt), "v"(gt) : "memory");
            }
        }
        asm volatile("s_wait_asynccnt 0x0" ::: "memory");
#endif
    } else {
        // Border tiles: guarded loads, zero-fill the pad region.
        for (int it = 0; it < 5; ++it) {
            const int i = tid + it * 256;
            if (i < NCELL) {
                const int hr = i / HALO_W;
                const int hc = i - hr * HALO_W;
                const int gy = y0 - 1 + hr;
                const int gx = x0 - 1 + hc;
                float vp = 0.0f, vt = 0.0f;
                if (gy >= 0 && gy < HW && gx >= 0 && gx < HW) {
                    const size_t g = base + (size_t)gy * HW + (size_t)gx;
                    vp = pred[g];
                    vt = target[g];
                }
                sP[i] = vp;
                sT[i] = vt;
            }
        }
    }
    __syncthreads();

    // Separable Sobel: horizontal pass into registers (6 halo rows/thread),
    // then vertical combine for 4 output rows.
    const int x  = tid & 63;           // output column in tile
    const int r0 = (tid >> 6) * 4;     // first output row (also first halo row)

    float aP[6], bP[6], aT[6], bT[6];
    #pragma unroll
    for (int k = 0; k < 6; ++k) {
        const float* rp = &sP[(r0 + k) * HALO_W + x];
        const float p0 = rp[0], p1 = rp[1], p2 = rp[2];
        aP[k] = p0 + 2.0f * p1 + p2;   // [1,2,1] horizontal
        bP[k] = p2 - p0;               // [-1,0,1] horizontal
        const float* rt = &sT[(r0 + k) * HALO_W + x];
        const float t0 = rt[0], t1 = rt[1], t2 = rt[2];
        aT[k] = t0 + 2.0f * t1 + t2;
        bT[k] = t2 - t0;
    }

    float acc = 0.0f;
    #pragma unroll
    for (int k = 0; k < 4; ++k) {
        const float exP = bP[k] + 2.0f * bP[k + 1] + bP[k + 2];
        const float eyP = aP[k + 2] - aP[k];
        const float mP  = __builtin_sqrtf(exP * exP + eyP * eyP + 1e-6f);
        const float exT = bT[k] + 2.0f * bT[k + 1] + bT[k + 2];
        const float eyT = aT[k + 2] - aT[k];
        const float mT  = __builtin_sqrtf(exT * exT + eyT * eyT + 1e-6f);
        acc += __builtin_fabsf(mP - mT);
    }

    // Deterministic intra-workgroup tree reduction.
    sred[tid] = acc;
    __syncthreads();
    #pragma unroll
    for (int off = 128; off > 0; off >>= 1) {
        if (tid < off) sred[tid] += sred[tid + off];
        __syncthreads();
    }
    if (tid == 0) partials[wg] = sred[0];
}

// Single-block, fixed-order final reduction -> bit-deterministic scalar mean.
__global__ __launch_bounds__(256) void reduce_partials_kernel(
    const float* __restrict__ partials, float* __restrict__ out,
    int n, float scale)
{
    __shared__ float s[256];
    float acc = 0.0f;
    for (int i = threadIdx.x; i < n; i += 256) acc += partials[i];
    s[threadIdx.x] = acc;
    __syncthreads();
    for (int off = 128; off > 0; off >>= 1) {
        if (threadIdx.x < off) s[threadIdx.x] += s[threadIdx.x + off];
        __syncthreads();
    }
    if (threadIdx.x == 0) out[0] = s[0] * scale;
}

extern "C" void kernel_launch(void* const* d_in, const int* in_sizes, int n_in,
                              void* d_out, int out_size, void* d_ws, size_t ws_size,
                              hipStream_t stream) {
    const float* pred   = (const float*)d_in[0];
    const float* target = (const float*)d_in[1];
    float* out      = (float*)d_out;
    float* partials = (float*)d_ws;

    const int n_img   = in_sizes[0] >> 18;       // elements / (512*512); = 48
    const int n_tiles = n_img * 256;             // 8 x 32 tiles per image

    sobel_edge_loss_kernel<<<n_tiles, 256, 0, stream>>>(pred, target, partials);

    const float scale = 1.0f / ((float)n_img * 262144.0f);
    reduce_partials_kernel<<<1, 256, 0, stream>>>(partials, out, n_tiles, scale);
}